// PointNetFeaturePropagation_33655363732106
// MI455X (gfx1250) — compile-verified
//
#include <hip/hip_runtime.h>
#include <hip/hip_bf16.h>

typedef __attribute__((ext_vector_type(2))) float v2f;
typedef __attribute__((ext_vector_type(8))) float v8f;

#define B_   8
#define N_   8192
#define S_   2048
#define D1_  128
#define D2_  256
#define C0_  384
#define C1_  256
#define C2_  128
#define BN_EPS_ 1e-5f

// ---------------- workspace layout (in floats) ----------------
#define WS_INTERP 0                      // 16777216 floats [B,D2,N]
#define WS_Y1     16777216               // 16777216 floats [B,C1,N]
#define WS_IDX    33554432               // 196608 ints   [B,N,3]
#define WS_WTS    33751040               // 196608 floats [B,N,3]
#define WS_STATS  33947648               // sum0[256] sq0[256] sum1[128] sq1[128]
#define WS_PRMS   33948416               // ac0[256] ac1[128] as interleaved (a,c) float2 pairs

// ============================================================
// 1) 3-NN: per (b,n) query, scan all S now-points staged in LDS
// ============================================================
__global__ __launch_bounds__(256) void knn_kernel(
    const float* __restrict__ xyz_sa, const float* __restrict__ xyz_now,
    int* __restrict__ idx, float* __restrict__ wts)
{
    __shared__ float sx[S_], sy[S_], sz[S_];
    const int b  = blockIdx.x >> 5;          // 32 blocks per batch
    const int n0 = (blockIdx.x & 31) << 8;   // 256 queries per block
    const float* xn = xyz_now + (size_t)b * 3 * S_;
    for (int i = threadIdx.x; i < S_; i += 256) {
        sx[i] = xn[i];
        sy[i] = xn[S_ + i];
        sz[i] = xn[2 * S_ + i];
    }
    __syncthreads();

    const int n = n0 + threadIdx.x;
    const float* xs = xyz_sa + (size_t)b * 3 * N_;
    const float qx = xs[n], qy = xs[N_ + n], qz = xs[2 * N_ + n];

    float d0 = 3.4e38f, d1 = 3.4e38f, d2 = 3.4e38f;
    int   i0 = 0, i1 = 0, i2 = 0;
    for (int s = 0; s < S_; ++s) {
        const float dx = sx[s] - qx, dy = sy[s] - qy, dz = sz[s] - qz;
        const float d  = fmaf(dx, dx, fmaf(dy, dy, dz * dz));
        if (d < d2) {
            if (d < d1) {
                d2 = d1; i2 = i1;
                if (d < d0) { d1 = d0; i1 = i0; d0 = d; i0 = s; }
                else        { d1 = d;  i1 = s; }
            } else { d2 = d; i2 = s; }
        }
    }
    const float r0 = 1.0f / (d0 + 1e-8f);
    const float r1 = 1.0f / (d1 + 1e-8f);
    const float r2 = 1.0f / (d2 + 1e-8f);
    const float inv = 1.0f / (r0 + r1 + r2);
    const size_t base = ((size_t)b * N_ + n) * 3;
    idx[base + 0] = i0; idx[base + 1] = i1; idx[base + 2] = i2;
    wts[base + 0] = r0 * inv; wts[base + 1] = r1 * inv; wts[base + 2] = r2 * inv;
}

// ============================================================
// 2) interp[b,c,n] = sum_k w_k * points_now[b,c,idx_k]
// ============================================================
__global__ __launch_bounds__(256) void interp_kernel(
    const float* __restrict__ pnow, const int* __restrict__ idx,
    const float* __restrict__ wts, float* __restrict__ interp)
{
    __shared__ int   si[64 * 3];
    __shared__ float sw[64 * 3];
    const int b  = blockIdx.x >> 7;           // N/64 = 128 blocks per batch
    const int n0 = (blockIdx.x & 127) << 6;
    const size_t base = ((size_t)b * N_ + n0) * 3;
    if (threadIdx.x < 192) {
        si[threadIdx.x] = idx[base + threadIdx.x];
        sw[threadIdx.x] = wts[base + threadIdx.x];
    }
    __syncthreads();

    const int nl = threadIdx.x & 63;
    const int cg = threadIdx.x >> 6;
    const int j0 = si[nl * 3 + 0], j1 = si[nl * 3 + 1], j2 = si[nl * 3 + 2];
    const float w0 = sw[nl * 3 + 0], w1 = sw[nl * 3 + 1], w2 = sw[nl * 3 + 2];
    const float* pb = pnow + (size_t)b * D2_ * S_;
    float* ob = interp + (size_t)b * D2_ * N_ + n0 + nl;
    for (int c = cg; c < D2_; c += 4) {
        const float* p = pb + (size_t)c * S_;
        ob[(size_t)c * N_] = fmaf(w0, p[j0], fmaf(w1, p[j1], w2 * p[j2]));
    }
}

// ============================================================
// 3) GEMM1: y1[b,o,n] = W0[o,:] . feat[b,:,n] + b0[o]
//    feat channel c<128 -> points_sa, c>=128 -> interp.
//    K loop split in two phases so each inner loop is branch-free.
//    One wave: 16 n-columns x 16 o-tiles (all 256 channels).
// ============================================================
__global__ __launch_bounds__(256) void gemm1_kernel(
    const float* __restrict__ psa, const float* __restrict__ interp,
    const float* __restrict__ W0, const float* __restrict__ b0,
    float* __restrict__ y1)
{
    const int b    = blockIdx.x >> 6;
    const int nblk = blockIdx.x & 63;
    const int wave = threadIdx.x >> 5;
    const int lane = threadIdx.x & 31;
    const int half = lane >> 4;
    const int lm   = lane & 15;
    const int n0   = nblk * 128 + wave * 16;

    // per-lane base pointers (k advances by constant stride)
    const float* pa = psa    + (size_t)b * D1_ * N_ + (size_t)(half * 2) * N_ + n0 + lm;
    const float* pi = interp + (size_t)b * D2_ * N_ + (size_t)(half * 2) * N_ + n0 + lm;
    const float* pw = W0 + lm * C0_ + half * 2;   // + t*16*C0 + k

    v8f acc[16];
#pragma unroll
    for (int t = 0; t < 16; ++t) acc[t] = v8f{0, 0, 0, 0, 0, 0, 0, 0};

    // phase 1: channels 0..127 from points_sa (32 K-steps)
    for (int s = 0; s < 32; ++s) {
        v2f bfrag;
        bfrag.x = pa[(size_t)(s * 4) * N_];
        bfrag.y = pa[(size_t)(s * 4 + 1) * N_];
#pragma unroll
        for (int t = 0; t < 16; ++t) {
            const v2f afrag = *(const v2f*)(pw + t * (16 * C0_) + s * 4);
            acc[t] = __builtin_amdgcn_wmma_f32_16x16x4_f32(
                false, afrag, false, bfrag, (short)0, acc[t], false, false);
        }
    }
    // phase 2: channels 128..383 from interp (64 K-steps)
    for (int s = 0; s < 64; ++s) {
        v2f bfrag;
        bfrag.x = pi[(size_t)(s * 4) * N_];
        bfrag.y = pi[(size_t)(s * 4 + 1) * N_];
#pragma unroll
        for (int t = 0; t < 16; ++t) {
            const v2f afrag = *(const v2f*)(pw + t * (16 * C0_) + 128 + s * 4);
            acc[t] = __builtin_amdgcn_wmma_f32_16x16x4_f32(
                false, afrag, false, bfrag, (short)0, acc[t], false, false);
        }
    }

    float* yb = y1 + (size_t)b * C1_ * N_ + n0 + lm;
#pragma unroll
    for (int t = 0; t < 16; ++t) {
#pragma unroll
        for (int j = 0; j < 8; ++j) {
            const int o = t * 16 + j + 8 * half;
            yb[(size_t)o * N_] = acc[t][j] + b0[o];
        }
    }
}

// ============================================================
// 4) per-channel sum / sumsq over (B, N). One block per channel,
//    coalesced float4 reads, LDS tree reduce. No atomics needed.
// ============================================================
__global__ __launch_bounds__(256) void stats_kernel(
    const float* __restrict__ y, float* __restrict__ sum, float* __restrict__ sq, int C)
{
    __shared__ float rs[256], rq[256];
    const int o = blockIdx.x;
    float s = 0.0f, q = 0.0f;
    for (int b = 0; b < B_; ++b) {
        const float4* p = (const float4*)(y + ((size_t)b * C + o) * N_);
        for (int i = threadIdx.x; i < N_ / 4; i += 256) {
            const float4 v = p[i];
            s += v.x + v.y + v.z + v.w;
            q += v.x * v.x + v.y * v.y + v.z * v.z + v.w * v.w;
        }
    }
    rs[threadIdx.x] = s;
    rq[threadIdx.x] = q;
    __syncthreads();
#pragma unroll
    for (int off = 128; off > 0; off >>= 1) {
        if (threadIdx.x < off) {
            rs[threadIdx.x] += rs[threadIdx.x + off];
            rq[threadIdx.x] += rq[threadIdx.x + off];
        }
        __syncthreads();
    }
    if (threadIdx.x == 0) { sum[o] = rs[0]; sq[o] = rq[0]; }
}

// ============================================================
// 5) fold BN stats into per-channel scale/shift (interleaved float2)
// ============================================================
__global__ void bnparams_kernel(
    const float* __restrict__ sum, const float* __restrict__ sq,
    const float* __restrict__ g, const float* __restrict__ be,
    float2* __restrict__ ac, int C, float invCount)
{
    const int i = threadIdx.x;
    if (i < C) {
        const float mean = sum[i] * invCount;
        const float var  = sq[i] * invCount - mean * mean;
        const float s    = g[i] * rsqrtf(var + BN_EPS_);
        ac[i] = make_float2(s, be[i] - mean * s);
    }
}

// ============================================================
// 6) GEMM2: y2[b,o,n] = W1[o,:] . relu(a0*y1 + c0) + b1[o]
//    BN0+ReLU applied on the fly; scale/shift pairs in LDS (b64 reads).
//    8 o-tiles per wave, K=256.
// ============================================================
__global__ __launch_bounds__(256) void gemm2_kernel(
    const float* __restrict__ y1, const float2* __restrict__ ac0,
    const float* __restrict__ W1, const float* __restrict__ b1,
    float* __restrict__ out)
{
    __shared__ float2 sAC[C1_];
    sAC[threadIdx.x] = ac0[threadIdx.x];
    __syncthreads();

    const int b    = blockIdx.x >> 6;
    const int nblk = blockIdx.x & 63;
    const int wave = threadIdx.x >> 5;
    const int lane = threadIdx.x & 31;
    const int half = lane >> 4;
    const int lm   = lane & 15;
    const int n0   = nblk * 128 + wave * 16;

    const float* py = y1 + (size_t)b * C1_ * N_ + (size_t)(half * 2) * N_ + n0 + lm;
    const float* pw = W1 + lm * C1_ + half * 2;   // + t*16*C1 + s*4

    v8f acc[8];
#pragma unroll
    for (int t = 0; t < 8; ++t) acc[t] = v8f{0, 0, 0, 0, 0, 0, 0, 0};

    const int kb = half * 2;
    for (int s = 0; s < 64; ++s) {
        const float2 p0 = sAC[s * 4 + kb];
        const float2 p1 = sAC[s * 4 + kb + 1];
        v2f bfrag;
        bfrag.x = fmaxf(fmaf(p0.x, py[(size_t)(s * 4) * N_],     p0.y), 0.0f);
        bfrag.y = fmaxf(fmaf(p1.x, py[(size_t)(s * 4 + 1) * N_], p1.y), 0.0f);
#pragma unroll
        for (int t = 0; t < 8; ++t) {
            const v2f afrag = *(const v2f*)(pw + t * (16 * C1_) + s * 4);
            acc[t] = __builtin_amdgcn_wmma_f32_16x16x4_f32(
                false, afrag, false, bfrag, (short)0, acc[t], false, false);
        }
    }

    float* ob = out + (size_t)b * C2_ * N_ + n0 + lm;
#pragma unroll
    for (int t = 0; t < 8; ++t) {
#pragma unroll
        for (int j = 0; j < 8; ++j) {
            const int o = t * 16 + j + 8 * half;
            ob[(size_t)o * N_] = acc[t][j] + b1[o];
        }
    }
}

// ============================================================
// 7) final BN+ReLU in place on d_out (float4 vectorized)
// ============================================================
__global__ __launch_bounds__(256) void finalize_kernel(
    float* __restrict__ out, const float2* __restrict__ ac1)
{
    const size_t i = ((size_t)blockIdx.x * 256 + threadIdx.x) * 4;
    const int ch = (int)((i >> 13) & (C2_ - 1));  // N = 8192 = 2^13
    const float2 p = ac1[ch];
    float4 v = *(float4*)(out + i);
    v.x = fmaxf(fmaf(p.x, v.x, p.y), 0.0f);
    v.y = fmaxf(fmaf(p.x, v.y, p.y), 0.0f);
    v.z = fmaxf(fmaf(p.x, v.z, p.y), 0.0f);
    v.w = fmaxf(fmaf(p.x, v.w, p.y), 0.0f);
    *(float4*)(out + i) = v;
}

// ============================================================
extern "C" void kernel_launch(void* const* d_in, const int* in_sizes, int n_in,
                              void* d_out, int out_size, void* d_ws, size_t ws_size,
                              hipStream_t stream)
{
    const float* xyz_sa   = (const float*)d_in[0];
    const float* xyz_now  = (const float*)d_in[1];
    const float* pts_sa   = (const float*)d_in[2];
    const float* pts_now  = (const float*)d_in[3];
    const float* W0       = (const float*)d_in[4];
    const float* b0       = (const float*)d_in[5];
    const float* g0       = (const float*)d_in[6];
    const float* be0      = (const float*)d_in[7];
    const float* W1       = (const float*)d_in[8];
    const float* b1       = (const float*)d_in[9];
    const float* g1       = (const float*)d_in[10];
    const float* be1      = (const float*)d_in[11];
    float* out = (float*)d_out;

    float*  ws     = (float*)d_ws;
    float*  interp = ws + WS_INTERP;
    float*  y1     = ws + WS_Y1;
    int*    idx    = (int*)(ws + WS_IDX);
    float*  wts    = ws + WS_WTS;
    float*  sum0   = ws + WS_STATS;
    float*  sq0    = sum0 + 256;
    float*  sum1   = sq0 + 256;
    float*  sq1    = sum1 + 128;
    float2* ac0    = (float2*)(ws + WS_PRMS);       // 256 pairs
    float2* ac1    = (float2*)(ws + WS_PRMS + 512); // 128 pairs

    const float invCount = 1.0f / (float)(B_ * N_);

    knn_kernel<<<B_ * (N_ / 256), 256, 0, stream>>>(xyz_sa, xyz_now, idx, wts);
    interp_kernel<<<B_ * (N_ / 64), 256, 0, stream>>>(pts_now, idx, wts, interp);
    gemm1_kernel<<<B_ * (N_ / 128), 256, 0, stream>>>(pts_sa, interp, W0, b0, y1);
    stats_kernel<<<C1_, 256, 0, stream>>>(y1, sum0, sq0, C1_);
    bnparams_kernel<<<1, 256, 0, stream>>>(sum0, sq0, g0, be0, ac0, C1_, invCount);
    gemm2_kernel<<<B_ * (N_ / 128), 256, 0, stream>>>(y1, ac0, W1, b1, out);
    stats_kernel<<<C2_, 256, 0, stream>>>(out, sum1, sq1, C2_);
    bnparams_kernel<<<1, 128, 0, stream>>>(sum1, sq1, g1, be1, ac1, C2_, invCount);
    finalize_kernel<<<(B_ * C2_ * N_) / (256 * 4), 256, 0, stream>>>(out, ac1);
}